// EmbeddedMLP_13108240187527
// MI455X (gfx1250) — compile-verified
//
#include <hip/hip_runtime.h>

typedef __attribute__((ext_vector_type(2))) float v2f;
typedef __attribute__((ext_vector_type(8))) float v8f;

#define B_TOT   2048
#define N_NODES 256
#define E_DIM   32
#define H_DIM   64
#define BT_ROWS 64      // b-rows per workgroup

// LDS row strides chosen so WMMA fragment loads are LDS-bank-conflict-free:
//  stride 68: bank = (4*row + col) % 64 -> 16 rows x col,col+2 groups disjoint
//  stride 72: bank = (8*row + col) % 64 -> rows k and k+2 hit disjoint 16-bank groups
#define XS_LD 68
#define H1_LD 68
#define W2_LD 72

// ---------------------------------------------------------------------------
// Precompute (tiny, once per call):
//   AT[i*256+j] = adjacency[j,i]                        (coalesced reload later)
//   c1[i,h] = emb[i]@W1_self + parent_emb[i]@W1_par + b1 (b-independent bias)
// ---------------------------------------------------------------------------
__global__ __launch_bounds__(64)
void precompute_kernel(const float* __restrict__ adj,
                       const float* __restrict__ emb,
                       const float* __restrict__ W1,
                       const float* __restrict__ b1,
                       float* __restrict__ c1,
                       float* __restrict__ AT) {
  const int i = blockIdx.x;
  const int t = threadIdx.x;
  __shared__ float red[64];
  __shared__ float pe[E_DIM];
  __shared__ float denom;

  for (int j = t; j < N_NODES; j += 64) AT[i * N_NODES + j] = adj[j * N_NODES + i];

  float s = 0.f;
  for (int j = t; j < N_NODES; j += 64) s += adj[j * N_NODES + i];
  red[t] = s;
  __syncthreads();
  if (t == 0) {
    float tot = 0.f;
    for (int k = 0; k < 64; k++) tot += red[k];
    denom = tot + 1e-8f;
  }
  __syncthreads();
  if (t < E_DIM) {
    float acc = 0.f;
    for (int j = 0; j < N_NODES; j++) acc += adj[j * N_NODES + i] * emb[j * E_DIM + t];
    pe[t] = acc / denom;
  }
  __syncthreads();
  float c = b1[t];
  for (int e = 0; e < E_DIM; e++) {
    c += emb[i * E_DIM + e] * W1[(N_NODES + e) * H_DIM + t];
    c += pe[e]              * W1[(N_NODES + E_DIM + e) * H_DIM + t];
  }
  c1[i * H_DIM + t] = c;
}

// ---------------------------------------------------------------------------
// Row-wise LayerNorm (var = E[x^2]-E[x]^2, eps=1e-5) + leaky-relu(0.2), in LDS.
// 64 rows, one thread per row (waves 0-1); negligible vs 160 WMMAs/wave.
// ---------------------------------------------------------------------------
__device__ __forceinline__ void ln_leaky_rows(float* buf, int ld,
                                              const float* g, const float* bt, int t) {
  if (t < 64) {
    float* row = buf + t * ld;
    float s = 0.f, s2 = 0.f;
#pragma unroll
    for (int k = 0; k < H_DIM; k++) { float v = row[k]; s += v; s2 += v * v; }
    const float m   = s  * (1.f / 64.f);
    const float var = s2 * (1.f / 64.f) - m * m;
    const float inv = rsqrtf(var + 1e-5f);
#pragma unroll
    for (int k = 0; k < H_DIM; k++) {
      float y = (row[k] - m) * inv * g[k] + bt[k];
      row[k] = (y >= 0.f) ? y : 0.2f * y;
    }
  }
}

// ---------------------------------------------------------------------------
// Fused kernel: one workgroup = (64 b-rows) x (1 node i), all 64 h columns.
//   GEMM1: (x-tile * A[:,i]) (64x256) @ W1[0:256,:] (256x64)  via v_wmma f32 16x16x4
//   LN+leaky -> GEMM2 (64x64)@(64x64) via v_wmma -> LN+leaky -> @W3 (64x2)
// ---------------------------------------------------------------------------
__global__ __launch_bounds__(256)
void fused_mlp_kernel(const float* __restrict__ x,
                      const float* __restrict__ W1,
                      const float* __restrict__ W2,
                      const float* __restrict__ b2,
                      const float* __restrict__ g1,  const float* __restrict__ bt1,
                      const float* __restrict__ g2,  const float* __restrict__ bt2,
                      const float* __restrict__ W3,  const float* __restrict__ b3,
                      const float* __restrict__ c1,  const float* __restrict__ AT,
                      float* __restrict__ out) {
  __shared__ float Xs [BT_ROWS * XS_LD];   // masked-x tile; reused as H2 buffer
  __shared__ float H1s[BT_ROWS * H1_LD];
  __shared__ float W2s[H_DIM * W2_LD];
  __shared__ float Acol[N_NODES];
  __shared__ float c1s[H_DIM], g1s[H_DIM], bt1s[H_DIM];
  __shared__ float g2s[H_DIM], bt2s[H_DIM], b2s[H_DIM];
  __shared__ float W3s[H_DIM * 2];
  __shared__ float b3s[2];

  const int tid = threadIdx.x;
  const int i   = blockIdx.y;
  const int b0  = blockIdx.x * BT_ROWS;

  // ---- stage block-invariant data ----
  Acol[tid] = AT[i * N_NODES + tid];            // coalesced (precomputed transpose)
  if (tid < H_DIM) {
    c1s[tid] = c1[i * H_DIM + tid];
    g1s[tid] = g1[tid];  bt1s[tid] = bt1[tid];
    g2s[tid] = g2[tid];  bt2s[tid] = bt2[tid];
    b2s[tid] = b2[tid];
  }
  if (tid < 2 * H_DIM) W3s[tid] = W3[tid];
  if (tid < 2)         b3s[tid] = b3[tid];
  for (int idx = tid; idx < H_DIM * H_DIM; idx += 256)
    W2s[(idx >> 6) * W2_LD + (idx & 63)] = W2[idx];
  __syncthreads();

  // ---- stage masked x tile: Xs[r][j] = x[b0+r, j] * A[j, i] ----
  for (int idx = tid; idx < BT_ROWS * N_NODES; idx += 256) {
    const int r = idx >> 8, c = idx & 255;
    Xs[r * XS_LD + c] = x[(b0 + r) * N_NODES + c] * Acol[c];
  }
  __syncthreads();

  // ---- wave tiling: 8 waves, 4x4 grid of 16x16 tiles, 2 h-tiles per wave ----
  const int lane = tid & 31;
  const int wave = tid >> 5;
  const int bt   = wave & 3;            // b-tile 0..3
  const int hp   = (wave >> 2) << 1;    // h-tile pair base: 0 or 2
  const int lrow = lane & 15;
  const int lhi  = lane >> 4;           // half-wave (K/M +offset group)
  const int arow = bt * 16 + lrow;      // A-frag row (M)
  const int n0   = hp * 16 + lrow;      // B/C-frag col, tile 0
  const int n1   = n0 + 16;             // tile 1

  // ---- GEMM1: B-fragments straight from global W1 (64 KB, cache-resident) ----
  v8f acc0 = {}, acc1 = {};
#pragma unroll 4
  for (int k0 = 0; k0 < N_NODES; k0 += 4) {
    const int ka = k0 + 2 * lhi;
    v2f a;   a.x   = Xs[arow * XS_LD + ka];      a.y   = Xs[arow * XS_LD + ka + 1];
    v2f bb0; bb0.x = W1[ka * H_DIM + n0];        bb0.y = W1[(ka + 1) * H_DIM + n0];
    v2f bb1; bb1.x = W1[ka * H_DIM + n1];        bb1.y = W1[(ka + 1) * H_DIM + n1];
    acc0 = __builtin_amdgcn_wmma_f32_16x16x4_f32(false, a, false, bb0, (short)0, acc0, false, false);
    acc1 = __builtin_amdgcn_wmma_f32_16x16x4_f32(false, a, false, bb1, (short)0, acc1, false, false);
  }
  {
    const float ca = c1s[n0], cb = c1s[n1];       // fused bias (emb terms + b1)
#pragma unroll
    for (int r = 0; r < 8; r++) {
      const int m = bt * 16 + 8 * lhi + r;        // C layout: VGPR r -> M = r (+8 hi half)
      H1s[m * H1_LD + n0] = acc0[r] + ca;
      H1s[m * H1_LD + n1] = acc1[r] + cb;
    }
  }
  __syncthreads();

  ln_leaky_rows(H1s, H1_LD, g1s, bt1s, tid);
  __syncthreads();

  // ---- GEMM2: h1 (64x64) @ W2 (LDS) ----
  v8f d0 = {}, d1 = {};
#pragma unroll
  for (int k0 = 0; k0 < H_DIM; k0 += 4) {
    const int ka = k0 + 2 * lhi;
    v2f a;   a.x   = H1s[arow * H1_LD + ka];     a.y   = H1s[arow * H1_LD + ka + 1];
    v2f bb0; bb0.x = W2s[ka * W2_LD + n0];       bb0.y = W2s[(ka + 1) * W2_LD + n0];
    v2f bb1; bb1.x = W2s[ka * W2_LD + n1];       bb1.y = W2s[(ka + 1) * W2_LD + n1];
    d0 = __builtin_amdgcn_wmma_f32_16x16x4_f32(false, a, false, bb0, (short)0, d0, false, false);
    d1 = __builtin_amdgcn_wmma_f32_16x16x4_f32(false, a, false, bb1, (short)0, d1, false, false);
  }
  {
    const float ba = b2s[n0], bb = b2s[n1];
#pragma unroll
    for (int r = 0; r < 8; r++) {
      const int m = bt * 16 + 8 * lhi + r;
      Xs[m * XS_LD + n0] = d0[r] + ba;            // Xs reused as H2 buffer
      Xs[m * XS_LD + n1] = d1[r] + bb;
    }
  }
  __syncthreads();

  ln_leaky_rows(Xs, XS_LD, g2s, bt2s, tid);
  __syncthreads();

  // ---- final: h2 @ W3 (64x2) + b3, write means / log_stds ----
  if (tid < BT_ROWS) {
    const float* row = Xs + tid * XS_LD;
    float mo = b3s[0], lo = b3s[1];
#pragma unroll
    for (int k = 0; k < H_DIM; k++) {
      const float v = row[k];
      mo += v * W3s[2 * k];
      lo += v * W3s[2 * k + 1];
    }
    const int b = b0 + tid;
    out[b * N_NODES + i]                   = mo;  // means
    out[B_TOT * N_NODES + b * N_NODES + i] = lo;  // log_stds
  }
}

// ---------------------------------------------------------------------------
extern "C" void kernel_launch(void* const* d_in, const int* in_sizes, int n_in,
                              void* d_out, int out_size, void* d_ws, size_t ws_size,
                              hipStream_t stream) {
  (void)in_sizes; (void)n_in; (void)out_size; (void)ws_size;
  const float* x   = (const float*)d_in[0];
  const float* adj = (const float*)d_in[1];
  const float* emb = (const float*)d_in[2];
  const float* W1  = (const float*)d_in[3];
  const float* b1  = (const float*)d_in[4];
  const float* g1  = (const float*)d_in[5];
  const float* bt1 = (const float*)d_in[6];
  const float* W2  = (const float*)d_in[7];
  const float* b2  = (const float*)d_in[8];
  const float* g2  = (const float*)d_in[9];
  const float* bt2 = (const float*)d_in[10];
  const float* W3  = (const float*)d_in[11];
  const float* b3  = (const float*)d_in[12];

  float* c1 = (float*)d_ws;                  // 256*64 floats  (64 KB)
  float* AT = c1 + N_NODES * H_DIM;          // 256*256 floats (256 KB)

  precompute_kernel<<<N_NODES, 64, 0, stream>>>(adj, emb, W1, b1, c1, AT);
  fused_mlp_kernel<<<dim3(B_TOT / BT_ROWS, N_NODES), 256, 0, stream>>>(
      x, W1, W2, b2, g1, bt1, g2, bt2, W3, b3, c1, AT, (float*)d_out);
}